// detector_32487132627105
// MI455X (gfx1250) — compile-verified
//
#include <hip/hip_runtime.h>
#include <stdint.h>

// Problem constants (match reference)
#define BB 32
#define NN 4096
#define CC 37
#define NC1 36
#define KK 100
#define DD 512
#define SCORE_TH 0.1f
#define NMS_TH 0.4f

// ---- TDM availability / arity handling -------------------------------------
#if defined(__has_builtin)
#if __has_builtin(__builtin_amdgcn_tensor_load_to_lds)
#define HAVE_TDM 1
#endif
#endif
#ifndef HAVE_TDM
#define HAVE_TDM 0
#endif

typedef unsigned int u32x4 __attribute__((ext_vector_type(4)));
typedef int          i32x4 __attribute__((ext_vector_type(4)));
typedef int          i32x8 __attribute__((ext_vector_type(8)));

// ---------------------------------------------------------------------------
// Kernel 1: masked, sorted top-K per (batch, class).
// Key packing: (ordered_uint(score) << 32) | (NN-1-n)  => max key == max score,
// ties broken by smallest n — identical ordering to jax.lax.top_k.
// ---------------------------------------------------------------------------
__device__ __forceinline__ uint32_t f32_ord(float f) {
  uint32_t u = __float_as_uint(f);
  return (u & 0x80000000u) ? ~u : (u | 0x80000000u);
}
__device__ __forceinline__ float f32_unord(uint32_t u) {
  u = (u & 0x80000000u) ? (u & 0x7FFFFFFFu) : ~u;
  return __uint_as_float(u);
}

__global__ __launch_bounds__(256) void topk_kernel(const float* __restrict__ scores,
                                                   int* __restrict__ out_idx,
                                                   float* __restrict__ out_val) {
  const int blk = blockIdx.x;          // 0 .. B*NC1-1
  const int b   = blk / NC1;
  const int cls = (blk % NC1) + 1;     // skip background class 0
  const int tid = threadIdx.x;

  // Each thread owns 16 candidates, n = j*256 + tid (coalesced across tid).
  unsigned long long loc[16];
#pragma unroll
  for (int j = 0; j < 16; ++j) {
    const int n = j * 256 + tid;
    float s = scores[((size_t)b * NN + n) * CC + cls];
    s = (s > SCORE_TH) ? s : -1.0f;    // masked score, exactly as reference
    loc[j] = ((unsigned long long)f32_ord(s) << 32) |
             (unsigned long long)(unsigned)(NN - 1 - n);
  }

  __shared__ unsigned long long wred[8];
  __shared__ unsigned long long bbest;
  const int lane = tid & 31;           // wave32
  const int wv   = tid >> 5;

  for (int it = 0; it < KK; ++it) {
    unsigned long long m = loc[0];
#pragma unroll
    for (int j = 1; j < 16; ++j) m = (loc[j] > m) ? loc[j] : m;
    // intra-wave (wave32) max reduction
#pragma unroll
    for (int off = 16; off > 0; off >>= 1) {
      unsigned long long o = __shfl_xor(m, off, 32);
      m = (o > m) ? o : m;
    }
    if (lane == 0) wred[wv] = m;
    __syncthreads();
    if (tid == 0) {
      unsigned long long bm = wred[0];
      for (int w = 1; w < 8; ++w) bm = (wred[w] > bm) ? wred[w] : bm;
      bbest = bm;
      out_idx[blk * KK + it] = NN - 1 - (int)(bm & 0xFFFFFFFFull);
      out_val[blk * KK + it] = f32_unord((uint32_t)(bm >> 32));
    }
    __syncthreads();
    const unsigned long long bm = bbest;
#pragma unroll
    for (int j = 0; j < 16; ++j)
      if (loc[j] == bm) loc[j] = 0ull;   // keys unique; 0 < any real key
  }
}

// ---------------------------------------------------------------------------
// TDM gather-mode descriptor: pull 16 rows (row length 512 f32, row stride 512)
// from a 4096x512 f32 tensor into LDS with a single tensor_load_to_lds.
// ---------------------------------------------------------------------------
#if HAVE_TDM
__device__ __forceinline__ void tdm_gather16(uint32_t lds_addr,
                                             unsigned long long gbase,
                                             const int* __restrict__ idx16) {
  u32x4 g0;
  g0.x = 0x80000001u;                                    // count=1, gather_mode=1, 16-bit idx
  g0.y = lds_addr;                                       // LDS destination (bytes)
  g0.z = (uint32_t)(gbase & 0xFFFFFFFFull);              // global_addr[31:0]
  g0.w = (uint32_t)((gbase >> 32) & 0x01FFFFFFull) | 0x80000000u;  // addr[56:32] | type=2

  i32x8 g1;
  g1[0] = 0x00020000;            // workgroup_mask=0, data_size=2 (4 bytes), no flags
  g1[1] = (int)(512u << 16);     // [63:48]  tensor_dim0 = 512
  g1[2] = (int)(4096u << 16);    // [95:80]  tensor_dim1 = 4096
  g1[3] = (int)(512u << 16);     // [127:112] tile_dim0  = 512
  g1[4] = 16;                    // [143:128] tile_dim1  = 16 valid gather indices
  g1[5] = 512;                   // [191:160] tensor_dim0_stride = 512
  g1[6] = 0;
  g1[7] = 0;

  i32x4 g2, g3;                  // 16-bit row indices, rows 0..7 / 8..15
#pragma unroll
  for (int q = 0; q < 4; ++q) {
    g2[q] = (idx16[2 * q]     & 0xFFFF) | (idx16[2 * q + 1] << 16);
    g3[q] = (idx16[2 * q + 8] & 0xFFFF) | (idx16[2 * q + 9] << 16);
  }
#if __clang_major__ >= 23
  i32x8 gz = {0, 0, 0, 0, 0, 0, 0, 0};
  __builtin_amdgcn_tensor_load_to_lds(g0, g1, g2, g3, gz, 0);
#else
  __builtin_amdgcn_tensor_load_to_lds(g0, g1, g2, g3, 0);
#endif
}
#endif

// ---------------------------------------------------------------------------
// Kernel 2: per (batch, class): TDM + async-DMA feature gather into LDS,
// overlapped with box decode + pairwise IoU + greedy NMS; emit 100x517 rows.
// ---------------------------------------------------------------------------
__global__ __launch_bounds__(256) void nms_kernel(const float* __restrict__ boxes,
                                                  const float* __restrict__ deltas,
                                                  const float* __restrict__ roi,
                                                  const float* __restrict__ im_info,
                                                  const int* __restrict__ tk_idx,
                                                  const float* __restrict__ tk_val,
                                                  float* __restrict__ out) {
  __shared__ float   s_feat[KK * DD];     // 200 KB, filled by TDM / async DMA
  __shared__ float4  s_box[KK];
  __shared__ float   s_area[KK];
  __shared__ float   s_val[KK];
  __shared__ int     s_idx[KK];
  __shared__ unsigned s_sup[KK][4];       // per-row 128-bit suppression mask (j>i only)
  __shared__ unsigned s_validm[4];

  const int blk = blockIdx.x;
  const int b   = blk / NC1;
  const int cls = (blk % NC1) + 1;
  const int tid = threadIdx.x;

  // ---- phase 1: load top-K results, clear masks --------------------------
  if (tid < KK) {
    s_idx[tid] = tk_idx[blk * KK + tid];
    s_val[tid] = tk_val[blk * KK + tid];
    s_sup[tid][0] = 0u; s_sup[tid][1] = 0u; s_sup[tid][2] = 0u; s_sup[tid][3] = 0u;
  }
  if (tid < 4) s_validm[tid] = 0u;
  __syncthreads();

  // ---- phase 2a: kick off DMA gather of the 100 feature rows --------------
  const float* rbase = roi + (size_t)b * NN * DD;
#if HAVE_TDM
  // 96 rows: six gather-mode TDM descriptors (16 rows each), wave 0 issues.
  if (tid < 32) {
    const uint32_t lds0 = (uint32_t)(uintptr_t)(&s_feat[0]);
    const unsigned long long gb = (unsigned long long)(uintptr_t)rbase;
    for (int ch = 0; ch < 6; ++ch)
      tdm_gather16(lds0 + (uint32_t)ch * 16u * DD * 4u, gb, &s_idx[ch * 16]);
  }
  const int tail0 = 96;   // rows 96..99 via per-lane async loads
#else
  const int tail0 = 0;    // no TDM builtin: all 100 rows via async loads
#endif
  for (int t = tid; t < (KK - tail0) * (DD / 4); t += 256) {
    const int k = tail0 + (t >> 7);        // row
    const int c = (t & 127) << 2;          // float column
    const unsigned long long ga =
        (unsigned long long)(uintptr_t)(rbase + (size_t)s_idx[k] * DD + c);
    const uint32_t lds = (uint32_t)(uintptr_t)(&s_feat[k * DD + c]);
    asm volatile("global_load_async_to_lds_b128 %0, %1, off"
                 :: "v"(lds), "v"(ga) : "memory");
  }

  // ---- phase 2b: decode + clip + scale the 100 candidate boxes -----------
  const float hmax = im_info[b * 3 + 0] - 1.0f;
  const float wmax = im_info[b * 3 + 1] - 1.0f;
  const float sc   = im_info[b * 3 + 2];
  if (tid < KK) {
    const int n = s_idx[tid];
    const float4 bb = ((const float4*)boxes)[(size_t)b * NN + n];
    const float4 dl = ((const float4*)deltas)[((size_t)b * NN + n) * CC + cls];
    const float w  = bb.z - bb.x + 1.0f;
    const float h  = bb.w - bb.y + 1.0f;
    const float cx = bb.x + 0.5f * w;
    const float cy = bb.y + 0.5f * h;
    const float pcx = (dl.x * 0.1f) * w + cx;
    const float pcy = (dl.y * 0.1f) * h + cy;
    const float pw  = __expf(dl.z * 0.2f) * w;
    const float ph  = __expf(dl.w * 0.2f) * h;
    float x1 = fminf(fmaxf(pcx - 0.5f * pw, 0.0f), wmax) / sc;
    float y1 = fminf(fmaxf(pcy - 0.5f * ph, 0.0f), hmax) / sc;
    float x2 = fminf(fmaxf(pcx + 0.5f * pw, 0.0f), wmax) / sc;
    float y2 = fminf(fmaxf(pcy + 0.5f * ph, 0.0f), hmax) / sc;
    s_box[tid]  = make_float4(x1, y1, x2, y2);
    s_area[tid] = (x2 - x1) * (y2 - y1);
    if (s_val[tid] > SCORE_TH) atomicOr(&s_validm[tid >> 5], 1u << (tid & 31));
  }
  __syncthreads();

  // ---- phase 3: pairwise IoU -> suppression bitmasks (j > i only) --------
  for (int p = tid; p < KK * KK; p += 256) {
    const int i = p / KK;
    const int j = p - i * KK;
    if (j > i) {
      const float4 bi = s_box[i], bj = s_box[j];
      const float ltx = fmaxf(bi.x, bj.x), lty = fmaxf(bi.y, bj.y);
      const float rbx = fminf(bi.z, bj.z), rby = fminf(bi.w, bj.w);
      const float iw = fmaxf(rbx - ltx, 0.0f), ih = fmaxf(rby - lty, 0.0f);
      const float inter = iw * ih;
      const float iou = inter / (s_area[i] + s_area[j] - inter + 1e-8f);
      if (iou > NMS_TH) atomicOr(&s_sup[i][j >> 5], 1u << (j & 31));
    }
  }
  __syncthreads();

  // ---- phase 4: greedy sequential NMS on 100-bit masks (redundant/thread) -
  unsigned km[4] = { s_validm[0], s_validm[1], s_validm[2], s_validm[3] };
  for (int i = 0; i < KK; ++i) {
    if ((km[i >> 5] >> (i & 31)) & 1u) {
      km[0] &= ~s_sup[i][0];
      km[1] &= ~s_sup[i][1];
      km[2] &= ~s_sup[i][2];
      km[3] &= ~s_sup[i][3];
    }
  }

  // ---- phase 5: drain both DMA paths, then emit output --------------------
#if HAVE_TDM
  __builtin_amdgcn_s_wait_tensorcnt(0);
#endif
  asm volatile("s_wait_asynccnt 0x0" ::: "memory");
  __syncthreads();   // make every wave's DMA'd LDS data visible block-wide

  // Output is write-once streaming (238 MB): non-temporal stores keep the
  // reused roi-feature slabs resident in the 192 MB L2.
  float* obase = out + (size_t)blk * KK * 517;
  for (int k = 0; k < KK; ++k) {
    const bool on = (km[k >> 5] >> (k & 31)) & 1u;
    const float4 bx = s_box[k];
    const float  v  = s_val[k];
    float* orow = obase + k * 517;
    for (int col = tid; col < 517; col += 256) {
      float x;
      if (col < 4)       x = (col == 0) ? bx.x : (col == 1) ? bx.y : (col == 2) ? bx.z : bx.w;
      else if (col == 4) x = v;
      else               x = s_feat[k * DD + (col - 5)];
      __builtin_nontemporal_store(on ? x : 0.0f, &orow[col]);
    }
  }
}

// ---------------------------------------------------------------------------
extern "C" void kernel_launch(void* const* d_in, const int* in_sizes, int n_in,
                              void* d_out, int out_size, void* d_ws, size_t ws_size,
                              hipStream_t stream) {
  (void)in_sizes; (void)n_in; (void)out_size; (void)ws_size;
  const float* boxes   = (const float*)d_in[0];
  const float* deltas  = (const float*)d_in[1];
  const float* scores  = (const float*)d_in[2];
  const float* roi     = (const float*)d_in[3];
  const float* im_info = (const float*)d_in[4];
  float* out = (float*)d_out;

  int*   tk_idx = (int*)d_ws;
  float* tk_val = (float*)((char*)d_ws + (size_t)BB * NC1 * KK * sizeof(int));

  const dim3 grid(BB * NC1);   // 1152 independent (batch, class) instances
  topk_kernel<<<grid, 256, 0, stream>>>(scores, tk_idx, tk_val);
  nms_kernel<<<grid, 256, 0, stream>>>(boxes, deltas, roi, im_info,
                                       tk_idx, tk_val, out);
}